// Model_84756884619935
// MI455X (gfx1250) — compile-verified
//
#include <hip/hip_runtime.h>
#include <hip/hip_bf16.h>

#define NNODES 50000
#define NEDGES 800000
#define NREL   16
#define DIM    128
#define OUTD   352   // 128 + 128 + 64 + 32

typedef __attribute__((ext_vector_type(2))) float v2f;
typedef __attribute__((ext_vector_type(8))) float v8f;

// ---------------------------------------------------------------------------
// fp32 WMMA: D(16x16,f32) = A(16x4,f32) * B(4x16,f32) + C
// ---------------------------------------------------------------------------
__device__ __forceinline__ v8f wmma_f32(v2f a, v2f b, v8f c) {
    return __builtin_amdgcn_wmma_f32_16x16x4_f32(false, a, false, b,
                                                 (short)0, c, false, false);
}

__device__ __forceinline__ float leaky(float x) { return x > 0.0f ? x : 0.01f * x; }

// Ordered-int trick for float atomic max (init buffer to -inf).
__device__ __forceinline__ void atomicMaxFloat(float* addr, float v) {
    if (!(v < 0.0f)) atomicMax(reinterpret_cast<int*>(addr), __float_as_int(v));
    else             atomicMin(reinterpret_cast<unsigned int*>(addr), __float_as_uint(v));
}

// ---------------------------------------------------------------------------
// h0 = entity_embed[node_ids]
// ---------------------------------------------------------------------------
__global__ void k_gather_h0(const float* __restrict__ ent, const int* __restrict__ ids,
                            float* __restrict__ h0, int total) {
    int i = blockIdx.x * blockDim.x + threadIdx.x;
    if (i >= total) return;
    int row = i / DIM, c = i - row * DIM;
    h0[i] = ent[(size_t)ids[row] * DIM + c];
}

__global__ void k_fill_neginf(float* __restrict__ p, int n) {
    int i = blockIdx.x * blockDim.x + threadIdx.x;
    if (i < n) p[i] = -__builtin_inff();
}

// ---------------------------------------------------------------------------
// proj[(n*NREL + r)*DIM + :] = h0[n,:] @ W_R[r]
// Block: 256 thr = 8 waves; each wave = 16 rows x 128 cols.  grid.y = relation.
// W_R[r] (64 KB) staged in LDS once per block; A fragments live in VGPRs
// (k-loop FULLY unrolled so the frag array promotes to registers, no scratch).
// ---------------------------------------------------------------------------
__global__ void __launch_bounds__(256)
k_proj_gemm(const float* __restrict__ h0, const float* __restrict__ W_R,
            float* __restrict__ proj, int n) {
    __shared__ float Wl[DIM * DIM];          // 64 KB of the WGP's 320 KB LDS

    const int r    = blockIdx.y;
    const int wave = threadIdx.x >> 5;
    const int lane = threadIdx.x & 31;
    const int half = lane >> 4;              // K-pair selector for A/B frags
    const int l16  = lane & 15;
    const int m0   = (blockIdx.x * 8 + wave) * 16;

    // cooperative LDS stage of W_R[r] (16384 floats, float4 per thread-iter)
    {
        const float4* Wg = reinterpret_cast<const float4*>(W_R + (size_t)r * DIM * DIM);
        float4*       Ws = reinterpret_cast<float4*>(Wl);
        for (int i = threadIdx.x; i < DIM * DIM / 4; i += 256) Ws[i] = Wg[i];
    }
    __syncthreads();

    // hoist all 32 A-fragments (64 floats/lane) — reused across all 8 n-tiles
    int rowA = m0 + l16; if (rowA >= n) rowA = n - 1;
    const float* Arow = h0 + (size_t)rowA * DIM;
    v2f afr[32];
#pragma unroll
    for (int t = 0; t < 32; ++t) {
        float2 av = *reinterpret_cast<const float2*>(Arow + t * 4 + half * 2);
        afr[t].x = av.x; afr[t].y = av.y;
    }

    const bool full = (m0 + 16 <= n);
    for (int nt = 0; nt < 8; ++nt) {
        const int n0 = nt * 16;
        v8f acc = {};
#pragma unroll          // FULL unroll: constant indices -> afr stays in VGPRs
        for (int t = 0; t < 32; ++t) {
            const int kk = t * 4 + half * 2;
            v2f b; b.x = Wl[kk * DIM + n0 + l16];
                   b.y = Wl[(kk + 1) * DIM + n0 + l16];
            acc = wmma_f32(afr[t], b, acc);
        }
        if (full) {
#pragma unroll
            for (int j = 0; j < 8; ++j)
                proj[((size_t)(m0 + j + half * 8) * NREL + r) * DIM + n0 + l16] = acc[j];
        } else {
#pragma unroll
            for (int j = 0; j < 8; ++j) {
                const int row = m0 + j + half * 8;
                if (row < n) proj[((size_t)row * NREL + r) * DIM + n0 + l16] = acc[j];
            }
        }
    }
}

// ---------------------------------------------------------------------------
// att[e] = sum_d t_r[d] * tanh(h_r[d] + e_r[d]);  amax[dst] = max(...)
// One wave (32 lanes) per edge, float4 per lane, wave32 butterfly reduce.
// ---------------------------------------------------------------------------
__global__ void k_att(const float* __restrict__ proj, const float* __restrict__ rel_e,
                      const int* __restrict__ src, const int* __restrict__ dst,
                      const int* __restrict__ rel, float* __restrict__ att,
                      float* __restrict__ amax, int e_cnt) {
    int e = (blockIdx.x * blockDim.x + threadIdx.x) >> 5;
    int lane = threadIdx.x & 31;
    if (e >= e_cnt) return;
    int s = src[e], d = dst[e], r = rel[e];
    float4 t4 = *reinterpret_cast<const float4*>(proj + ((size_t)s * NREL + r) * DIM + lane * 4);
    float4 h4 = *reinterpret_cast<const float4*>(proj + ((size_t)d * NREL + r) * DIM + lane * 4);
    float4 r4 = *reinterpret_cast<const float4*>(rel_e + (size_t)r * DIM + lane * 4);
    float p = t4.x * tanhf(h4.x + r4.x) + t4.y * tanhf(h4.y + r4.y)
            + t4.z * tanhf(h4.z + r4.z) + t4.w * tanhf(h4.w + r4.w);
#pragma unroll
    for (int o = 16; o > 0; o >>= 1) p += __shfl_xor(p, o, 32);
    if (lane == 0) { att[e] = p; atomicMaxFloat(&amax[d], p); }
}

// ea[e] = exp(att - amax[dst]); denom[dst] += ea
__global__ void k_ea(const float* __restrict__ att, const float* __restrict__ amax,
                     const int* __restrict__ dst, float* __restrict__ ea,
                     float* __restrict__ denom, int e_cnt) {
    int e = blockIdx.x * blockDim.x + threadIdx.x;
    if (e >= e_cnt) return;
    int d = dst[e];
    float v = expf(att[e] - amax[d]);
    ea[e] = v;
    atomicAdd(&denom[d], v);
}

// hn[dst,:] += h[src,:] * (ea/denom[dst]).  One wave per edge, float4 per lane.
__global__ void k_scatter(const float* __restrict__ h, const float* __restrict__ ea,
                          const float* __restrict__ denom, const int* __restrict__ src,
                          const int* __restrict__ dst, float* __restrict__ hn, int e_cnt) {
    int e = (blockIdx.x * blockDim.x + threadIdx.x) >> 5;
    int lane = threadIdx.x & 31;
    if (e >= e_cnt) return;
    int s = src[e], d = dst[e];
    float w = ea[e] / denom[d];
    float4 hv = *reinterpret_cast<const float4*>(h + (size_t)s * DIM + lane * 4);
    float* base = hn + (size_t)d * DIM + lane * 4;
    atomicAdd(base + 0, hv.x * w);
    atomicAdd(base + 1, hv.y * w);
    atomicAdd(base + 2, hv.z * w);
    atomicAdd(base + 3, hv.w * w);
}

// ---------------------------------------------------------------------------
// raw = leaky((h+hn) @ Wa) + leaky((h*hn) @ Wb)   Wa,Wb: [DIM, K]
// A fragments (h+hn, h*hn) hoisted into VGPRs; k-loop FULLY unrolled.
// ---------------------------------------------------------------------------
__global__ void __launch_bounds__(256)
k_conv_gemm(const float* __restrict__ h, const float* __restrict__ hn,
            const float* __restrict__ Wa, const float* __restrict__ Wb,
            float* __restrict__ raw, int n, int K) {
    const int wave = threadIdx.x >> 5;
    const int lane = threadIdx.x & 31;
    const int half = lane >> 4;
    const int l16  = lane & 15;
    const int m0   = (blockIdx.x * 8 + wave) * 16;

    int rowA = m0 + l16; if (rowA >= n) rowA = n - 1;
    const float* hrow  = h  + (size_t)rowA * DIM;
    const float* hnrow = hn + (size_t)rowA * DIM;

    // hoist A fragments for both GEMMs (aA = h+hn, aM = h*hn)
    v2f aA[32], aM[32];
#pragma unroll
    for (int t = 0; t < 32; ++t) {
        const int kk = t * 4 + half * 2;
        float2 hv = *reinterpret_cast<const float2*>(hrow  + kk);
        float2 nv = *reinterpret_cast<const float2*>(hnrow + kk);
        aA[t].x = hv.x + nv.x; aA[t].y = hv.y + nv.y;
        aM[t].x = hv.x * nv.x; aM[t].y = hv.y * nv.y;
    }

    const bool full = (m0 + 16 <= n);
    const int ntiles = K >> 4;
    for (int nt = 0; nt < ntiles; ++nt) {
        const int n0 = nt * 16;
        v8f accA = {}, accB = {};
#pragma unroll          // FULL unroll: keeps aA/aM in registers (no scratch)
        for (int t = 0; t < 32; ++t) {
            const int kk = t * 4 + half * 2;
            v2f bA; bA.x = Wa[kk * K + n0 + l16]; bA.y = Wa[(kk + 1) * K + n0 + l16];
            v2f bB; bB.x = Wb[kk * K + n0 + l16]; bB.y = Wb[(kk + 1) * K + n0 + l16];
            accA = wmma_f32(aA[t], bA, accA);
            accB = wmma_f32(aM[t], bB, accB);
        }
        if (full) {
#pragma unroll
            for (int j = 0; j < 8; ++j)
                raw[(size_t)(m0 + j + half * 8) * K + n0 + l16]
                    = leaky(accA[j]) + leaky(accB[j]);
        } else {
#pragma unroll
            for (int j = 0; j < 8; ++j) {
                const int row = m0 + j + half * 8;
                if (row < n) raw[(size_t)row * K + n0 + l16]
                                 = leaky(accA[j]) + leaky(accB[j]);
            }
        }
    }
}

// out[row, off:off+K] = raw[row,:] / max(||raw[row,:]||, 1e-12). Wave per row.
__global__ void k_norm_out(const float* __restrict__ raw, float* __restrict__ out,
                           int n, int K, int off) {
    int row  = (blockIdx.x * blockDim.x + threadIdx.x) >> 5;
    int lane = threadIdx.x & 31;
    if (row >= n) return;
    float s = 0.0f;
    for (int k = lane; k < K; k += 32) { float x = raw[(size_t)row * K + k]; s += x * x; }
#pragma unroll
    for (int o = 16; o > 0; o >>= 1) s += __shfl_xor(s, o, 32);
    float inv = 1.0f / fmaxf(sqrtf(s), 1e-12f);
    for (int k = lane; k < K; k += 32)
        out[(size_t)row * OUTD + off + k] = raw[(size_t)row * K + k] * inv;
}

// out[:, 0:128] = h0
__global__ void k_copy_h0(const float* __restrict__ h0, float* __restrict__ out, int total) {
    int i = blockIdx.x * blockDim.x + threadIdx.x;
    if (i >= total) return;
    int row = i / DIM, c = i - row * DIM;
    out[(size_t)row * OUTD + c] = h0[i];
}

// ---------------------------------------------------------------------------
extern "C" void kernel_launch(void* const* d_in, const int* in_sizes, int n_in,
                              void* d_out, int out_size, void* d_ws, size_t ws_size,
                              hipStream_t stream) {
    const float* ent   = (const float*)d_in[0];
    const float* rel_e = (const float*)d_in[1];
    const float* W_R   = (const float*)d_in[2];
    const float* Wlist_a[3] = { (const float*)d_in[3], (const float*)d_in[5], (const float*)d_in[7] };
    const float* Wlist_b[3] = { (const float*)d_in[4], (const float*)d_in[6], (const float*)d_in[8] };
    const int* node_ids = (const int*)d_in[9];
    const int* rel_ids  = (const int*)d_in[10];
    const int* src      = (const int*)d_in[11];
    const int* dst      = (const int*)d_in[12];
    float* out = (float*)d_out;

    // workspace layout (floats)
    float* ws = (float*)d_ws;
    size_t o = 0;
    float* h0    = ws + o; o += (size_t)NNODES * DIM;
    float* proj  = ws + o; o += (size_t)NNODES * NREL * DIM;
    float* att   = ws + o; o += NEDGES;
    float* amax  = ws + o; o += NNODES;
    float* ea    = ws + o; o += NEDGES;
    float* denom = ws + o; o += NNODES;
    float* hnA   = ws + o; o += (size_t)NNODES * DIM;
    float* hnB   = ws + o; o += (size_t)NNODES * DIM;
    float* raw   = ws + o; o += (size_t)NNODES * DIM;

    const int ND = NNODES * DIM;

    k_gather_h0<<<(ND + 255) / 256, 256, 0, stream>>>(ent, node_ids, h0, ND);
    k_fill_neginf<<<(NNODES + 255) / 256, 256, 0, stream>>>(amax, NNODES);
    hipMemsetAsync(denom, 0, NNODES * sizeof(float), stream);

    dim3 gproj((NNODES + 127) / 128, NREL);
    k_proj_gemm<<<gproj, 256, 0, stream>>>(h0, W_R, proj, NNODES);

    k_att<<<(NEDGES + 7) / 8, 256, 0, stream>>>(proj, rel_e, src, dst, rel_ids, att, amax, NEDGES);
    k_ea<<<(NEDGES + 255) / 256, 256, 0, stream>>>(att, amax, dst, ea, denom, NEDGES);

    const int Ks[3]   = { 128, 64, 32 };
    const int offs[3] = { 128, 256, 320 };
    const float* hcur = h0;
    float* hbuf[2] = { hnA, hnB };

    for (int l = 0; l < 3; ++l) {
        float* hn = hbuf[l & 1];
        hipMemsetAsync(hn, 0, (size_t)ND * sizeof(float), stream);
        k_scatter<<<(NEDGES + 7) / 8, 256, 0, stream>>>(hcur, ea, denom, src, dst, hn, NEDGES);
        k_conv_gemm<<<(NNODES + 127) / 128, 256, 0, stream>>>(hcur, hn, Wlist_a[l], Wlist_b[l],
                                                              raw, NNODES, Ks[l]);
        k_norm_out<<<(NNODES + 7) / 8, 256, 0, stream>>>(raw, out, NNODES, Ks[l], offs[l]);
        hcur = hn;
    }
    k_copy_h0<<<(ND + 255) / 256, 256, 0, stream>>>(h0, out, ND);
}